// XPINN_1915555414464
// MI455X (gfx1250) — compile-verified
//
#include <hip/hip_runtime.h>
#include <hip/hip_bf16.h>

// ---------------------------------------------------------------------------
// XPINN loss on MI455X (gfx1250).
// Closed-form Taylor-jet propagation through the tanh MLPs replaces JAX's
// forward-over-reverse AD. Every jet component shares the same 128x128 matmul
// per layer -> fp32 WMMA (V_WMMA_F32_16X16X4_F32), wave32, 16-point tiles.
// Layer weights are staged TRANSPOSED into LDS so every WMMA B-fragment is a
// single immediate-offset ds_load_b64; activation jets are double-buffered in
// LDS; next layer's weights are prefetched (global_prefetch) during compute.
// Deterministic: fixed per-block workspace slots + fixed-order final reduce.
// ---------------------------------------------------------------------------

typedef float v2f __attribute__((ext_vector_type(2)));
typedef float v4f __attribute__((ext_vector_type(4)));
typedef float v8f __attribute__((ext_vector_type(8)));

#define H     128
#define ROWS  132   // act LDS row pad (floats): bank spread, 8B aligned
#define RK    130   // transposed-weight LDS row pad (floats)
#define WSZ   (H * RK)            // floats of LDS for one staged layer
#define ASZ(nc) ((nc) * 16 * ROWS) // floats of LDS for one act buffer

static __device__ inline v8f wmma_f32(v2f a, v2f b, v8f c) {
  // D(16x16,f32) += A(16x4,f32) x B(4x16,f32)
  return __builtin_amdgcn_wmma_f32_16x16x4_f32(
      /*neg_a=*/false, a, /*neg_b=*/false, b,
      /*c_mod=*/(short)0, c, /*reuse_a=*/false, /*reuse_b=*/false);
}

struct NetParams {
  const float* Win;   // [2][H]
  const float* bi;    // [H]
  const float* Wh;    // [3][H][H]
  const float* bh;    // [3][H]
  const float* Wout;  // [H]
  const float* bo;    // [1]
};

// tanh jet composition. comp order: 0:v 1:x 2:y 3:xx 4:yy 5:xy 6:xxx 7:xxy 8:xyy 9:yyy
template <int NCOMP>
static __device__ inline void tanh_jets(const float* z, float* a) {
  float t  = tanhf(z[0]);
  float s1 = 1.0f - t * t;
  a[0] = t;
  if constexpr (NCOMP > 1) {
    float zx = z[1], zy = z[2], zxx = z[3], zyy = z[4];
    float s2 = -2.0f * t * s1;
    a[1] = s1 * zx;
    a[2] = s1 * zy;
    a[3] = fmaf(s2, zx * zx, s1 * zxx);
    a[4] = fmaf(s2, zy * zy, s1 * zyy);
    if constexpr (NCOMP == 10) {
      float zxy = z[5], zxxx = z[6], zxxy = z[7], zxyy = z[8], zyyy = z[9];
      float s3 = -2.0f * s1 * s1 - 2.0f * t * s2;
      a[5] = fmaf(s2, zx * zy, s1 * zxy);
      a[6] = s3 * zx * zx * zx + 3.0f * s2 * zx * zxx + s1 * zxxx;
      a[7] = s3 * zx * zx * zy + s2 * (zxx * zy + 2.0f * zx * zxy) + s1 * zxxy;
      a[8] = s3 * zx * zy * zy + s2 * (zyy * zx + 2.0f * zy * zxy) + s1 * zxyy;
      a[9] = s3 * zy * zy * zy + 3.0f * s2 * zy * zyy + s1 * zyyy;
    }
  }
}

static __device__ inline float reduce16(float v) {
  v += __shfl_xor(v, 1, 32);
  v += __shfl_xor(v, 2, 32);
  v += __shfl_xor(v, 4, 32);
  v += __shfl_xor(v, 8, 32);
  return v;  // each 16-group leader holds its 16-lane sum
}

// Cooperatively stage W (row-major [k][n]) into LDS transposed: wlds[n*RK+k].
// Coalesced b128 global reads, bank-spread b32 LDS scatter stores.
static __device__ inline void stage_weightsT(float* wlds, const float* __restrict__ W) {
  const int tid = threadIdx.x;
  const int nt  = blockDim.x;
  for (int idx = tid; idx < (H * H) / 4; idx += nt) {
    int k  = idx >> 5;          // 32 quads per 128-wide row
    int n4 = (idx & 31) * 4;
    v4f w = *(const v4f*)(W + k * H + n4);
    wlds[(n4 + 0) * RK + k] = w.x;
    wlds[(n4 + 1) * RK + k] = w.y;
    wlds[(n4 + 2) * RK + k] = w.z;
    wlds[(n4 + 3) * RK + k] = w.w;
  }
}

// One wave computes NCOMP output jets for 16 points (xs/ys already tile-based).
// wlds: block-shared staged-weight LDS (WSZ floats).
// act0/act1: per-wave LDS jet buffers, each ASZ(NCOMP) floats.
template <int NCOMP>
static __device__ void mlp_jets(float* wlds, float* act0, float* act1,
                                NetParams P,
                                const float* __restrict__ xs,
                                const float* __restrict__ ys,
                                float* out /* NCOMP per-lane */) {
  const int lane  = threadIdx.x & 31;
  const int half  = lane >> 4;   // 0: lanes 0-15, 1: lanes 16-31
  const int mlane = lane & 15;

  // ---- input layer: z = x*Win0 + y*Win1 + bi; jets: zx=Win0, zy=Win1 ----
  for (int it = 0; it < (16 * H) / 32; ++it) {
    int idx = it * 32 + lane;
    int m = idx >> 7;          // point in tile
    int n = idx & (H - 1);     // hidden unit
    float x = xs[m], y = ys[m];
    float wx = P.Win[n], wy = P.Win[H + n];
    float z[NCOMP], a[NCOMP];
#pragma unroll
    for (int c = 0; c < NCOMP; ++c) z[c] = 0.0f;
    z[0] = fmaf(x, wx, fmaf(y, wy, P.bi[n]));
    if constexpr (NCOMP > 1) { z[1] = wx; z[2] = wy; }
    tanh_jets<NCOMP>(z, a);
#pragma unroll
    for (int c = 0; c < NCOMP; ++c) act0[(c * 16 + m) * ROWS + n] = a[c];
  }

  // ---- 3 hidden layers: stage W^T to LDS, WMMA jet matmuls, tanh jets ----
  float* ain = act0;
  float* aout = act1;
  for (int layer = 0; layer < 3; ++layer) {
    stage_weightsT(wlds, P.Wh + layer * H * H);
    __syncthreads();  // wlds ready (also orders act0 fill / prev aout)

    // prefetch next layer's weights into cache while we compute
    if (layer < 2) {
      const float* Wn = P.Wh + (layer + 1) * H * H;
      for (int i = threadIdx.x * 32; i < H * H; i += blockDim.x * 32)
        __builtin_prefetch(Wn + i, 0, 0);
    }

    const float* bb = P.bh + layer * H;
    for (int nt = 0; nt < 8; ++nt) {
      v8f acc[NCOMP];
#pragma unroll
      for (int c = 0; c < NCOMP; ++c)
#pragma unroll
        for (int i = 0; i < 8; ++i) acc[c][i] = 0.0f;

      const int n = nt * 16 + mlane;          // this lane's output column
      const float* brow = wlds + n * RK;      // staged W^T row for column n
      for (int kt = 0; kt < 32; ++kt) {
        int k = kt * 4 + 2 * half;
        v2f bf = *(const v2f*)(brow + k);     // {W[k][n], W[k+1][n]} : 1 ds_load_b64
#pragma unroll
        for (int c = 0; c < NCOMP; ++c) {
          v2f af = *(const v2f*)(ain + (c * 16 + mlane) * ROWS + k);
          acc[c] = wmma_f32(af, bf, acc[c]);
        }
      }
      float bn = bb[n];
#pragma unroll
      for (int i = 0; i < 8; ++i) {
        float z[NCOMP], a[NCOMP];
        z[0] = acc[0][i] + bn;
#pragma unroll
        for (int c = 1; c < NCOMP; ++c) z[c] = acc[c][i];
        tanh_jets<NCOMP>(z, a);
        int M = i + 8 * half;  // C-layout: VGPR i, half -> row
#pragma unroll
        for (int c = 0; c < NCOMP; ++c) aout[(c * 16 + M) * ROWS + n] = a[c];
      }
    }
    __syncthreads();  // aout complete; wlds free for next layer's staging
    float* t = ain; ain = aout; aout = t;
  }

  // ---- output layer: dot(h_jet, Wout) + bo (value only) ----
  float s[NCOMP];
#pragma unroll
  for (int c = 0; c < NCOMP; ++c) s[c] = 0.0f;
  for (int n = half * 64; n < half * 64 + 64; ++n) {
    float w = P.Wout[n];
#pragma unroll
    for (int c = 0; c < NCOMP; ++c)
      s[c] = fmaf(ain[(c * 16 + mlane) * ROWS + n], w, s[c]);
  }
#pragma unroll
  for (int c = 0; c < NCOMP; ++c) s[c] += __shfl_xor(s[c], 16, 32);
  s[0] += P.bo[0];
#pragma unroll
  for (int c = 0; c < NCOMP; ++c) out[c] = s[c];
}

// ---------------- boundary: 4 waves x 16 pts, NCOMP=1 ----------------
#define NB_B 128   // 2 nets * 4096/64
__global__ void __launch_bounds__(128)
kBoundary(const float* xb1, const float* yb1, const float* ub1,
          const float* xb2, const float* yb2, const float* ub2,
          NetParams P1, NetParams P2, float* ws) {
  extern __shared__ float smem[];
  float* wlds = smem;
  int wave = threadIdx.x >> 5;
  float* a0 = smem + WSZ + wave * 2 * ASZ(1);
  float* a1 = a0 + ASZ(1);

  int net  = (blockIdx.x >= 64);
  int b    = blockIdx.x & 63;
  int base = b * 64 + wave * 16;
  const float* xs = (net ? xb2 : xb1) + base;
  const float* ys = (net ? yb2 : yb1) + base;
  const float* us = (net ? ub2 : ub1) + base;
  NetParams P = net ? P2 : P1;

  float out[1];
  mlp_jets<1>(wlds, a0, a1, P, xs, ys, out);

  float e  = us[threadIdx.x & 15] - out[0];
  float sq = reduce16(e * e);

  __shared__ float wsumB[4];
  if ((threadIdx.x & 31) == 0) wsumB[wave] = sq;
  __syncthreads();
  if (threadIdx.x == 0)
    ws[blockIdx.x] = wsumB[0] + wsumB[1] + wsumB[2] + wsumB[3];
}

// ---------------- residual: 2 waves x 16 pts, NCOMP=5 ----------------
#define NB_F 2048  // 2 nets * 32768/32
__global__ void __launch_bounds__(64)
kResidual(const float* xf1, const float* yf1, const float* ff1,
          const float* xf2, const float* yf2, const float* ff2,
          NetParams P1, NetParams P2, float* ws) {
  extern __shared__ float smem[];
  float* wlds = smem;
  int wave = threadIdx.x >> 5;
  float* a0 = smem + WSZ + wave * 2 * ASZ(5);
  float* a1 = a0 + ASZ(5);

  int net  = (blockIdx.x >= 1024);
  int b    = blockIdx.x & 1023;
  int base = b * 32 + wave * 16;
  const float* xs = (net ? xf2 : xf1) + base;
  const float* ys = (net ? yf2 : yf1) + base;
  const float* fs = (net ? ff2 : ff1) + base;
  NetParams P = net ? P2 : P1;

  float out[5];
  mlp_jets<5>(wlds, a0, a1, P, xs, ys, out);

  float f  = out[3] + out[4] - fs[threadIdx.x & 15];
  float sq = reduce16(f * f);

  __shared__ float wsumR[2];
  if ((threadIdx.x & 31) == 0) wsumR[wave] = sq;
  __syncthreads();
  if (threadIdx.x == 0) ws[blockIdx.x] = wsumR[0] + wsumR[1];
}

// ---------------- interface: 1 wave x 16 pts, NCOMP=10, both nets ----------------
#define NB_I 256   // 4096/16
__global__ void __launch_bounds__(32)
kInterface(const float* xi, const float* yi, const float* fi,
           const float* iw, NetParams P1, NetParams P2, float* ws) {
  extern __shared__ float smem[];
  float* wlds = smem;
  float* a0 = smem + WSZ;
  float* a1 = a0 + ASZ(10);

  int base = blockIdx.x * 16;
  float o1[10], o2[10];
  mlp_jets<10>(wlds, a0, a1, P1, xi + base, yi + base, o1);
  __syncthreads();
  mlp_jets<10>(wlds, a0, a1, P2, xi + base, yi + base, o2);

  float fsrc = fi[base + (threadIdx.x & 15)];
  float f1   = o1[3] + o1[4] - fsrc;
  float f2   = o2[3] + o2[4] - fsrc;
  float g1x  = o1[6] + o1[8], g1y = o1[7] + o1[9];  // grad of laplacian, net1
  float g2x  = o2[6] + o2[8], g2y = o2[7] + o2[9];
  float du   = o1[0] - o2[0];
  float dfd  = f1 - f2;
  float dfy  = o1[2] - o2[2];
  float dyy  = o1[4] - o2[4];
  float dx   = o1[1] - o2[1];
  float dxx  = o1[3] - o2[3];

  // mse_i_uavg == mse_i_u / 2 algebraically -> fold weights
  float term = iw[2] * (f1 * f1 + f2 * f2)
             + iw[3] * dfd * dfd
             + (iw[0] + 0.5f * iw[1]) * du * du
             + iw[4] * dfy * dfy
             + iw[6] * (g1x * g1x + g1y * g1y + g2x * g2x + g2y * g2y)
             + iw[5] * dyy * dyy
             + iw[7] * dx * dx
             + iw[8] * dxx * dxx;

  float sq = reduce16(term);
  if (threadIdx.x == 0) ws[blockIdx.x] = sq;
}

// ---------------- final deterministic reduction ----------------
__global__ void __launch_bounds__(32)
kFinal(const float* ws, float* out) {
  int lane = threadIdx.x;
  float sb = 0.0f, sf = 0.0f, si = 0.0f;
  for (int i = lane; i < NB_B; i += 32) sb += ws[i];
  for (int i = lane; i < NB_F; i += 32) sf += ws[NB_B + i];
  for (int i = lane; i < NB_I; i += 32) si += ws[NB_B + NB_F + i];
#pragma unroll
  for (int m = 1; m < 32; m <<= 1) {
    sb += __shfl_xor(sb, m, 32);
    sf += __shfl_xor(sf, m, 32);
    si += __shfl_xor(si, m, 32);
  }
  if (lane == 0)
    out[0] = 20.0f * sb / 4096.0f + sf / 32768.0f + 5.0f * si / 4096.0f;
}

extern "C" void kernel_launch(void* const* d_in, const int* in_sizes, int n_in,
                              void* d_out, int out_size, void* d_ws, size_t ws_size,
                              hipStream_t stream) {
  const float* xb1 = (const float*)d_in[0];
  const float* yb1 = (const float*)d_in[1];
  const float* ub1 = (const float*)d_in[2];
  const float* xb2 = (const float*)d_in[3];
  const float* yb2 = (const float*)d_in[4];
  const float* ub2 = (const float*)d_in[5];
  const float* xf1 = (const float*)d_in[6];
  const float* yf1 = (const float*)d_in[7];
  const float* ff1 = (const float*)d_in[8];
  const float* xf2 = (const float*)d_in[9];
  const float* yf2 = (const float*)d_in[10];
  const float* ff2 = (const float*)d_in[11];
  const float* xi1 = (const float*)d_in[12];
  const float* yi1 = (const float*)d_in[13];
  const float* fi1 = (const float*)d_in[14];
  const float* iw  = (const float*)d_in[15];

  NetParams P1{(const float*)d_in[16], (const float*)d_in[17],
               (const float*)d_in[18], (const float*)d_in[19],
               (const float*)d_in[20], (const float*)d_in[21]};
  NetParams P2{(const float*)d_in[22], (const float*)d_in[23],
               (const float*)d_in[24], (const float*)d_in[25],
               (const float*)d_in[26], (const float*)d_in[27]};

  float* ws = (float*)d_ws;  // needs (128+2048+256)*4 = 9728 bytes

  const size_t LDS_B = (WSZ + 4u * 2u * ASZ(1))  * sizeof(float);  // ~131 KB
  const size_t LDS_R = (WSZ + 2u * 2u * ASZ(5))  * sizeof(float);  // ~230 KB
  const size_t LDS_I = (WSZ + 2u * ASZ(10))      * sizeof(float);  // ~230 KB

  kBoundary <<<NB_B, 128, LDS_B, stream>>>(xb1, yb1, ub1, xb2, yb2, ub2, P1, P2, ws);
  kResidual <<<NB_F,  64, LDS_R, stream>>>(xf1, yf1, ff1, xf2, yf2, ff2, P1, P2, ws + NB_B);
  kInterface<<<NB_I,  32, LDS_I, stream>>>(xi1, yi1, fi1, iw, P1, P2, ws + NB_B + NB_F);
  kFinal    <<<1,     32, 0,     stream>>>(ws, (float*)d_out);
}